// GenericPool_25503515803832
// MI455X (gfx1250) — compile-verified
//
#include <hip/hip_runtime.h>
#include <stdint.h>

// Segment-mean pooling for MI455X (gfx1250, wave32).
// sums/counts live in d_ws (5.12MB + 40KB -> resident in 192MB L2).
// Input stream uses CDNA5 async global->LDS DMA double-buffering (ASYNCcnt);
// output stream uses non-temporal stores so the means table stays in L2.

typedef float v4f __attribute__((ext_vector_type(4)));

// Pointer types matching __builtin_amdgcn_global_load_async_to_lds_b128's
// signature: (v4i addrspace(1)*, v4i addrspace(3)*, imm offset, imm cpol).
typedef int v4i __attribute__((vector_size(16)));
typedef __attribute__((address_space(1))) v4i* global_v4i_ptr;
typedef __attribute__((address_space(3))) v4i* shared_v4i_ptr;
// Volatile AS(3) view of the staging buffer: forces a real ds_load_b128 that
// cannot be folded away (the async DMA's LDS write is invisible to the
// optimizer's memory model).
typedef volatile __attribute__((address_space(3))) v4f* shared_v4f_vptr;

#define SEGS 10000
#define DIM  128

// ---------------- CDNA5 async global->LDS copy (16B per lane) ----------------
__device__ __forceinline__ void async_copy_b128(const void* gsrc, void* lds_dst) {
#if __has_builtin(__builtin_amdgcn_global_load_async_to_lds_b128)
  __builtin_amdgcn_global_load_async_to_lds_b128(
      (global_v4i_ptr)gsrc, (shared_v4i_ptr)lds_dst,
      /*imm offset=*/0, /*cpol=*/0);
#else
  unsigned lofs = (unsigned)(uintptr_t)(shared_v4i_ptr)lds_dst;
  asm volatile("global_load_async_to_lds_b128 %0, %1, off"
               :: "v"(lofs), "v"(gsrc) : "memory");
#endif
}

#if __has_builtin(__builtin_amdgcn_s_wait_asynccnt)
#define WAIT_ASYNC(n) do { __builtin_amdgcn_s_wait_asynccnt(n); \
                           asm volatile("" ::: "memory"); } while (0)
#else
#define WAIT_ASYNC(n) asm volatile("s_wait_asynccnt " #n ::: "memory")
#endif

// ---------------- kernel 1: zero sums+counts ---------------------------------
__global__ __launch_bounds__(256) void k_zero(float* __restrict__ p, int n) {
  int i = blockIdx.x * 256 + threadIdx.x;
  if (i < n) p[i] = 0.0f;
}

// ---------------- kernel 2: atomic segment accumulate ------------------------
// One wave per row: lane L owns floats [4L, 4L+3] of the 128-wide row.
// Per-wave LDS double buffer fed by async DMA; atomics issue while the next
// row's 512B tile is already streaming into LDS.
__global__ __launch_bounds__(256) void k_accum(const v4f* __restrict__ in4,
                                               const int* __restrict__ idx,
                                               float* __restrict__ sums,
                                               float* __restrict__ counts,
                                               int nrows) {
  __shared__ v4f buf[8][2][32];              // 8 waves x 2 buffers x 512B = 8KB
  const int lane = threadIdx.x & 31;
  const int wv   = threadIdx.x >> 5;
  const int w    = blockIdx.x * 8 + wv;      // global wave id
  const int W    = gridDim.x * 8;            // total waves

  int row = w;
  if (row < nrows) async_copy_b128(in4 + (size_t)row * 32 + lane, &buf[wv][0][lane]);
  int cur = 0;

  for (; row < nrows; row += W) {
    const int next = row + W;
    if (next < nrows) {
      async_copy_b128(in4 + (size_t)next * 32 + lane, &buf[wv][cur ^ 1][lane]);
      WAIT_ASYNC(1);                         // current tile done, next in flight
    } else {
      WAIT_ASYNC(0);
    }

    // Forced ds_load_b128 from the DMA-filled buffer (volatile AS3 view).
    const v4f v = *(shared_v4f_vptr)&buf[wv][cur][lane];

    const int seg = idx[row];                // lane-uniform broadcast load
    float* dst = sums + (size_t)seg * DIM + lane * 4;
    __hip_atomic_fetch_add(dst + 0, v.x, __ATOMIC_RELAXED, __HIP_MEMORY_SCOPE_AGENT);
    __hip_atomic_fetch_add(dst + 1, v.y, __ATOMIC_RELAXED, __HIP_MEMORY_SCOPE_AGENT);
    __hip_atomic_fetch_add(dst + 2, v.z, __ATOMIC_RELAXED, __HIP_MEMORY_SCOPE_AGENT);
    __hip_atomic_fetch_add(dst + 3, v.w, __ATOMIC_RELAXED, __HIP_MEMORY_SCOPE_AGENT);
    if (lane == 0)
      __hip_atomic_fetch_add(counts + seg, 1.0f, __ATOMIC_RELAXED, __HIP_MEMORY_SCOPE_AGENT);
    cur ^= 1;
  }
}

// ---------------- kernel 3: sums -> means (in place) -------------------------
__global__ __launch_bounds__(256) void k_means(float* __restrict__ sums,
                                               const float* __restrict__ counts,
                                               int n) {
  int i = blockIdx.x * 256 + threadIdx.x;
  if (i < n) {
    const float c = counts[i >> 7];          // i / DIM
    sums[i] *= 1.0f / fmaxf(c, 1.0f);
  }
}

// ---------------- kernel 4: gather means back to rows ------------------------
// Means table (5.12MB) stays hot in L2; the 512MB output stream is written
// with non-temporal hints so it does not evict it.
__global__ __launch_bounds__(256) void k_gather(const float* __restrict__ means,
                                                const int* __restrict__ idx,
                                                v4f* __restrict__ out4,
                                                int nrows) {
  const int lane = threadIdx.x & 31;
  const int w    = blockIdx.x * 8 + (threadIdx.x >> 5);
  const int W    = gridDim.x * 8;

  for (int row = w; row < nrows; row += W) {
    if (row + W < nrows) __builtin_prefetch(idx + row + W, 0, 0);  // global_prefetch_b8
    const int seg = idx[row];
    const v4f v = *((const v4f*)(means + (size_t)seg * DIM) + lane);
    __builtin_nontemporal_store(v, out4 + (size_t)row * 32 + lane);
  }
}

// ---------------- host-side launch -------------------------------------------
extern "C" void kernel_launch(void* const* d_in, const int* in_sizes, int n_in,
                              void* d_out, int out_size, void* d_ws, size_t ws_size,
                              hipStream_t stream) {
  const float* input = (const float*)d_in[0];   // [N,128] f32
  const int*   index = (const int*)d_in[1];     // [N] int
  float* out    = (float*)d_out;                // [N,128] f32
  float* sums   = (float*)d_ws;                 // [SEGS*DIM] f32
  float* counts = sums + (size_t)SEGS * DIM;    // [SEGS] f32
  const int nrows = in_sizes[1];                // N = 1,000,000

  const int nz = SEGS * DIM + SEGS;
  k_zero  <<<(nz + 255) / 256, 256, 0, stream>>>(sums, nz);
  k_accum <<<2048, 256, 0, stream>>>((const v4f*)input, index, sums, counts, nrows);
  k_means <<<(SEGS * DIM + 255) / 256, 256, 0, stream>>>(sums, counts, SEGS * DIM);
  k_gather<<<2048, 256, 0, stream>>>(sums, index, (v4f*)out, nrows);
}